// RBFMIXTURELQR_58445914964042
// MI455X (gfx1250) — compile-verified
//
#include <hip/hip_runtime.h>

typedef __attribute__((ext_vector_type(2))) float v2f;
typedef __attribute__((ext_vector_type(8))) float v8f;

#define NRBF 11
#define THREADS1 256
#define ITEMS 8
#define NPB (THREADS1 * ITEMS)   // 2048 positions per block

__device__ __forceinline__ float sigmoid_clip(float x) {
  x = fminf(fmaxf(x, -10.f), 10.f);
  return 1.f / (1.f + __expf(-x));
}

// Kernel 1: elementwise u, write u to output, WMMA-based block-local inclusive
// scan of u (per channel) into workspace, block totals into partials.
__global__ __launch_bounds__(THREADS1) void RBF_k1_u_localscan(
    const float2* __restrict__ yin, const float2* __restrict__ xa,
    const float2* __restrict__ xb, const float* __restrict__ rbfw_raw,
    const float* __restrict__ weights, const float* __restrict__ g1r,
    const float* __restrict__ g2r, float2* __restrict__ u_out,
    float* __restrict__ ysl0, float* __restrict__ ysl1,
    float* __restrict__ partials, int N, float inv_lm1, int nblk)
{
  __shared__ float ldsU[2][NPB];
  __shared__ float waveTot[2][8];

  const int tid  = threadIdx.x;
  const int lane = tid & 31;
  const int wv   = tid >> 5;
  const long blockBase = (long)blockIdx.x * NPB;

  const float gain1 = __expf(g1r[0]);
  const float gain2 = __expf(g2r[0]);
  const float RBFw  = sigmoid_clip(rbfw_raw[0]);
  const float wdt   = RBFw * 0.2f;             // RBFw*(tmax-tmin)/(NRBF-1) = RBFw*2/10
  const float inv2w2 = 1.f / (2.f * wdt * wdt);

  float wreg[NRBF];
#pragma unroll
  for (int j = 0; j < NRBF; ++j) wreg[j] = weights[j];

  // ---- elementwise u, stash per-channel into LDS (zero-padded past N) ----
#pragma unroll
  for (int j = 0; j < ITEMS; ++j) {
    int local = j * THREADS1 + tid;
    long pos = blockBase + local;
    float ux = 0.f, uy = 0.f;
    if (pos < N) {
      float2 y = yin[pos];
      float2 a = xa[pos];
      float2 b = xb[pos];
      float t = (float)pos * (2.f * inv_lm1) - 1.f;   // t in [-1,1]
      float acc = 0.f;
#pragma unroll
      for (int r = 0; r < NRBF; ++r) {
        float d = t - (-1.f + 0.2f * (float)r);       // centers linspace(-1,1,11)
        acc = fmaf(wreg[r], __expf(-d * d * inv2w2), acc);
      }
      float w = sigmoid_clip(acc);
      float u1x = -gain1 * (y.x - a.x), u1y = -gain1 * (y.y - a.y);
      float u2x = -gain2 * (y.x - b.x), u2y = -gain2 * (y.y - b.y);
      ux = u1x * (1.f - w) + u2x * w;
      uy = u1y * (1.f - w) + u2y * w;
      u_out[pos] = make_float2(ux, uy);
    }
    ldsU[0][local] = ux;
    ldsU[1][local] = uy;
  }
  __syncthreads();

  // ---- WMMA scan: D = L(16x16, lower-tri ones) @ X(16x16 tile) in FP32 ----
  // Each wave scans 256 elements/channel: 16 segments (cols) x 16 elems (rows).
  // L split into four 16x4 K-groups, X into four 4x16 K-groups; 4 chained WMMAs.
  const int n    = lane & 15;   // A row M == B col N == segment id
  const int half = lane >> 4;

  v8f accv[2];
#pragma unroll
  for (int c = 0; c < 2; ++c) {
    v8f acc = {0.f, 0.f, 0.f, 0.f, 0.f, 0.f, 0.f, 0.f};
    const float* base = &ldsU[c][wv * 256];
#pragma unroll
    for (int g = 0; g < 4; ++g) {
      int k0 = 4 * g + 2 * half;                 // K covered by this lane-half
      v2f A;                                     // A[m,k] = (k <= m) lower-tri ones
      A.x = (k0 <= n) ? 1.f : 0.f;
      A.y = ((k0 + 1) <= n) ? 1.f : 0.f;
      v2f B;                                     // B[k,n] = X[4g+k, n] = elem n*16+4g+k
      B.x = base[n * 16 + k0];
      B.y = base[n * 16 + k0 + 1];
      acc = __builtin_amdgcn_wmma_f32_16x16x4_f32(false, A, false, B,
                                                  (short)0, acc, false, false);
    }
    // cross-segment offsets: column totals live in VGPR7 of lanes 16..31
    float tot  = __shfl(acc[7], 16 + n, 32);
    float incl = tot;
#pragma unroll
    for (int d = 1; d < 16; d <<= 1) {
      float o = __shfl_up(incl, (unsigned)d, 16);
      if (n >= d) incl += o;
    }
    float off = incl - tot;                      // exclusive offset of segment n
#pragma unroll
    for (int v = 0; v < 8; ++v) acc[v] += off;
    float wtot = __shfl(acc[7], 31, 32);         // tile total (element 255)
    if (lane == 0) waveTot[c][wv] = wtot;
    accv[c] = acc;
  }
  __syncthreads();

  // ---- add wave offsets, write local inclusive scan (contiguous b128 runs) ----
#pragma unroll
  for (int c = 0; c < 2; ++c) {
    float woff = 0.f;
    for (int q = 0; q < wv; ++q) woff += waveTot[c][q];
    v8f acc = accv[c];
#pragma unroll
    for (int v = 0; v < 8; ++v) acc[v] += woff;
    // lane holds rows m = v + 8*half of column n -> elements n*16+8*half+v (contiguous)
    float* dst = (c == 0 ? ysl0 : ysl1) + blockBase + wv * 256 + n * 16 + half * 8;
#pragma unroll
    for (int v = 0; v < 8; ++v) dst[v] = acc[v];
  }
  if (tid == 0) {
#pragma unroll
    for (int c = 0; c < 2; ++c) {
      float s = 0.f;
#pragma unroll
      for (int q = 0; q < 8; ++q) s += waveTot[c][q];
      partials[c * nblk + blockIdx.x] = s;
    }
  }
}

// Kernel 2: exclusive scan of per-block totals (P <= 512 per channel).
__global__ __launch_bounds__(1024) void RBF_k2_scan_partials(float* partials, int P)
{
  __shared__ float s[1024];
  int tid = threadIdx.x;
  int c = tid >> 9;
  int i = tid & 511;
  float own = (i < P) ? partials[c * P + i] : 0.f;
  s[tid] = own;
  __syncthreads();
  for (int d = 1; d < 512; d <<= 1) {
    float t = (i >= d) ? s[tid - d] : 0.f;
    __syncthreads();
    s[tid] += t;
    __syncthreads();
  }
  if (i < P) partials[c * P + i] = s[tid] - own;   // exclusive
}

// Kernel 3: y[k] = yin[0] + (1/60) * (local_scan + block_offset); y[0] = yin[0].
__global__ __launch_bounds__(256) void RBF_k3_apply(
    const float* __restrict__ ysl0, const float* __restrict__ ysl1,
    const float* __restrict__ partials, const float2* __restrict__ yin,
    float2* __restrict__ y_out, long N, int nblk)
{
  long idx = (long)blockIdx.x * blockDim.x + threadIdx.x;
  long stride = (long)gridDim.x * blockDim.x;
  float2 y0 = yin[0];
  const float Bp = 1.f / 60.f;
  if (idx == 0) y_out[0] = y0;
  for (long pos = idx; pos < N; pos += stride) {
    int b = (int)(pos >> 11);
    float vx = ysl0[pos] + partials[b];
    float vy = ysl1[pos] + partials[nblk + b];
    y_out[pos + 1] = make_float2(fmaf(Bp, vx, y0.x), fmaf(Bp, vy, y0.y));
  }
}

extern "C" void kernel_launch(void* const* d_in, const int* in_sizes, int n_in,
                              void* d_out, int out_size, void* d_ws, size_t ws_size,
                              hipStream_t stream)
{
  const int leny = in_sizes[0] / 2;       // yin is (LENY, 2)
  const int N = leny - 1;                 // scan length
  const int nblk = (N + NPB - 1) / NPB;   // 512 for LENY = 1<<20
  const float inv_lm1 = 1.0f / (float)(leny - 1);

  const float2* yin = (const float2*)d_in[0];
  const float2* xa  = (const float2*)d_in[1];
  const float2* xb  = (const float2*)d_in[2];
  const float*  rbf = (const float*)d_in[3];
  const float*  wts = (const float*)d_in[4];
  const float*  g1  = (const float*)d_in[5];
  const float*  g2  = (const float*)d_in[6];

  float2* y_out = (float2*)d_out;                                  // LENY rows
  float2* u_out = (float2*)((float*)d_out + 2 * (long)leny);       // LENY-1 rows

  float* wsf = (float*)d_ws;
  long plane = (long)nblk * NPB;          // zero-risk padded plane per channel
  float* ysl0 = wsf;
  float* ysl1 = wsf + plane;
  float* partials = wsf + 2 * plane;      // 2*nblk floats

  RBF_k1_u_localscan<<<nblk, THREADS1, 0, stream>>>(
      yin, xa, xb, rbf, wts, g1, g2, u_out, ysl0, ysl1, partials,
      N, inv_lm1, nblk);
  RBF_k2_scan_partials<<<1, 1024, 0, stream>>>(partials, nblk);
  RBF_k3_apply<<<1024, 256, 0, stream>>>(ysl0, ysl1, partials, yin, y_out,
                                         (long)N, nblk);
}